// TopKLayer_56667798503660
// MI455X (gfx1250) — compile-verified
//
#include <hip/hip_runtime.h>

// TopK spatial mask, faithful to the reference's rank-vs-index comparison.
// One 256-thread workgroup per (n,c) row of 3136 fp32 elements.
//   1) async-copy row global -> LDS (CDNA5 GLOBAL_LOAD_ASYNC_TO_LDS_B128 path)
//   2) build 64-bit keys (|x| bits << 12 | spatial idx), pad to 4096 with max keys
//   3) in-LDS bitonic sort (ascending)  -> sorted position == ascending rank
//   4) thr_idx = original index at sorted position hw-k  (index of k-th largest)
//   5) vectorized masked writeback: out[j] = x[j] * (rank[j] < thr_idx)

typedef unsigned long long u64;
typedef unsigned int u32;

#define HW      3136
#define HW4     (HW / 4)     // 784 float4 per row
#define SORT_N  4096
#define TOPK_K  313
#define THREADS 256

#if defined(__has_builtin)
#if __has_builtin(__builtin_amdgcn_global_load_async_to_lds_b32)
#define HAVE_ASYNC_B32 1
#endif
#if __has_builtin(__builtin_amdgcn_global_load_async_to_lds_b128)
#define HAVE_ASYNC_B128 1
#endif
#if __has_builtin(__builtin_amdgcn_s_wait_asynccnt)
#define HAVE_WAIT_ASYNC 1
#endif
#endif
#ifndef HAVE_ASYNC_B32
#define HAVE_ASYNC_B32 0
#endif
#ifndef HAVE_ASYNC_B128
#define HAVE_ASYNC_B128 0
#endif
#ifndef HAVE_WAIT_ASYNC
#define HAVE_WAIT_ASYNC 0
#endif

// Address-space-qualified pointer types for the async builtins
// (global-AS src first, LDS-AS dst second). b32 takes int*, b128 takes
// a 16-byte int vector* per hipcc's diagnostics.
typedef int v4i __attribute__((vector_size(16)));
typedef __attribute__((address_space(1))) int g_int;
typedef __attribute__((address_space(3))) int l_int;
typedef __attribute__((address_space(1))) v4i g_v4i;
typedef __attribute__((address_space(3))) v4i l_v4i;

__global__ __launch_bounds__(THREADS) void topk_hw_mask_kernel(
    const float* __restrict__ x, float* __restrict__ out) {
  __shared__ float vals[HW];       // staged row
  __shared__ u64   skey[SORT_N];   // sort keys: (|x|bits << 12) | idx
  __shared__ u32   rankOf[HW];     // ascending rank by spatial index
  __shared__ int   thrIdx;         // spatial index of k-th largest |x|

  const int tid = threadIdx.x;
  const long long base = (long long)blockIdx.x * HW;
  const float* src = x + base;

  // ---- Stage row into LDS (wide async DMA-style copy) --------------------
#if HAVE_ASYNC_B128
  for (int j = tid; j < HW4; j += THREADS) {
    __builtin_amdgcn_global_load_async_to_lds_b128(
        (g_v4i*)(src + 4 * j),
        (l_v4i*)(&vals[4 * j]),
        /*offset=*/0, /*cpol=*/0);
  }
#elif HAVE_ASYNC_B32
  for (int j = tid; j < HW; j += THREADS) {
    __builtin_amdgcn_global_load_async_to_lds_b32(
        (g_int*)(src + j),
        (l_int*)(&vals[j]),
        /*offset=*/0, /*cpol=*/0);
  }
#else
  for (int j = tid; j < HW; j += THREADS) vals[j] = src[j];
#endif
#if HAVE_ASYNC_B128 || HAVE_ASYNC_B32
#if HAVE_WAIT_ASYNC
  __builtin_amdgcn_s_wait_asynccnt(0);
#else
  asm volatile("s_wait_asynccnt 0" ::: "memory");
#endif
#endif
  __syncthreads();

  // ---- Build keys; pad [HW, SORT_N) with max keys ------------------------
  for (int j = tid; j < SORT_N; j += THREADS) {
    if (j < HW) {
      u32 bits = __float_as_uint(vals[j]);
      u64 ab   = (u64)(bits & 0x7fffffffu);   // |x| bits: monotone for abs fp32
      skey[j]  = (ab << 12) | (u32)j;         // stable tie-break by index
    } else {
      skey[j] = ~0ull;                        // sorts to the end
    }
  }
  __syncthreads();

  // ---- Bitonic sort ascending (4096 keys, 2048 c/e per pass) -------------
  for (int size = 2; size <= SORT_N; size <<= 1) {
    for (int stride = size >> 1; stride > 0; stride >>= 1) {
      for (int t = tid; t < (SORT_N / 2); t += THREADS) {
        int lo = (t << 1) - (t & (stride - 1));
        int hi = lo + stride;
        bool asc = ((lo & size) == 0);
        u64 a = skey[lo];
        u64 b = skey[hi];
        if ((a > b) == asc) { skey[lo] = b; skey[hi] = a; }
      }
      __syncthreads();
    }
  }

  // ---- Scatter ranks; capture threshold spatial index --------------------
  // Ascending position p IS the ascending rank (pads live in [HW, SORT_N)).
  // k-th largest (descending, ties measure-zero for fp32 normals) sits at
  // ascending position HW - K; thr_idx is its ORIGINAL spatial index.
  for (int p = tid; p < HW; p += THREADS) {
    u64 key = skey[p];
    int idx = (int)(key & 0xfffu);
    rankOf[idx] = (u32)p;
    if (p == (HW - TOPK_K)) thrIdx = idx;
  }
  __syncthreads();

  // ---- Vectorized masked writeback (float4 -> global_store_b128) ---------
  const int ti = thrIdx;
  float4* out4 = reinterpret_cast<float4*>(out + base);
  for (int q = tid; q < HW4; q += THREADS) {
    const int j = 4 * q;
    float4 v;
    v.x = ((int)rankOf[j + 0] < ti) ? vals[j + 0] : 0.0f;
    v.y = ((int)rankOf[j + 1] < ti) ? vals[j + 1] : 0.0f;
    v.z = ((int)rankOf[j + 2] < ti) ? vals[j + 2] : 0.0f;
    v.w = ((int)rankOf[j + 3] < ti) ? vals[j + 3] : 0.0f;
    out4[q] = v;
  }
}

extern "C" void kernel_launch(void* const* d_in, const int* in_sizes, int n_in,
                              void* d_out, int out_size, void* d_ws, size_t ws_size,
                              hipStream_t stream) {
  (void)n_in; (void)d_ws; (void)ws_size; (void)out_size;
  const float* x = (const float*)d_in[0];
  float* out = (float*)d_out;
  const int rows = in_sizes[0] / HW;  // 32*256 = 8192
  topk_hw_mask_kernel<<<dim3(rows), dim3(THREADS), 0, stream>>>(x, out);
}